// BiologicalMemory_55499567398930
// MI455X (gfx1250) — compile-verified
//
#include <hip/hip_runtime.h>
#include <hip/hip_bf16.h>
#include <stdint.h>

typedef __attribute__((ext_vector_type(2))) float v2f;
typedef __attribute__((ext_vector_type(8))) float v8f;

#define DIM   1024
#define SEQ   2048
#define NMEM  131072
#define EPSV  1e-8f
#define THRESH 0.65f

// ---- monotonic float <-> uint mapping for packed-u64 argmax -------------
__device__ __forceinline__ unsigned int f32_to_ordered(float f) {
  unsigned int b = __float_as_uint(f);
  return (b & 0x80000000u) ? ~b : (b | 0x80000000u);
}
__device__ __forceinline__ float ordered_to_f32(unsigned int u) {
  unsigned int b = (u & 0x80000000u) ? (u & 0x7FFFFFFFu) : ~u;
  return __uint_as_float(b);
}

// ---- K1: partial column sums of query (8 row-chunks), init argmax slot --
// ws layout: part[8*1024] floats, then u64 best-key at byte offset 32768.
__global__ void k_partial_q(const float* __restrict__ query,
                            float* __restrict__ part,
                            unsigned long long* __restrict__ best) {
  int t = blockIdx.x * blockDim.x + threadIdx.x;   // 8192 threads
  if (t == 0) *best = 0ull;
  int d     = t & (DIM - 1);
  int chunk = t >> 10;                              // 0..7
  const int RPC = SEQ / 8;                          // 256 rows per chunk
  const float* p = query + (size_t)chunk * RPC * DIM + d;
  float s = 0.f;
  #pragma unroll 4
  for (int r = 0; r < RPC; ++r) s += p[(size_t)r * DIM];
  part[t] = s;                                      // un-normalized (cosine is scale-invariant)
}

// ---- K2: HBM-bound streamer: cosine sim per bank row + global argmax ----
__global__ void __launch_bounds__(256) k_sims(const float* __restrict__ bank,
                                              const float* __restrict__ part,
                                              unsigned long long* __restrict__ best) {
  const int lane = threadIdx.x & 31;
  const int wib  = threadIdx.x >> 5;                        // wave in block (0..7)
  const int wid  = blockIdx.x * (blockDim.x >> 5) + wib;
  const int nw   = gridDim.x * (blockDim.x >> 5);

  // Preload q into 8 float4 regs per lane (sum the 8 partial chunks once).
  float4 qv[8];
  const float4* pp = (const float4*)part;
  #pragma unroll
  for (int j = 0; j < 8; ++j) {
    int b4 = lane + 32 * j;                                 // float4 index in [0,256)
    float4 a = pp[b4];
    #pragma unroll
    for (int c = 1; c < 8; ++c) {
      float4 b = pp[c * 256 + b4];
      a.x += b.x; a.y += b.y; a.z += b.z; a.w += b.w;
    }
    qv[j] = a;
  }
  // ||q|| (wave butterfly reduce)
  float qn = 0.f;
  #pragma unroll
  for (int j = 0; j < 8; ++j)
    qn = fmaf(qv[j].x, qv[j].x, fmaf(qv[j].y, qv[j].y,
         fmaf(qv[j].z, qv[j].z, fmaf(qv[j].w, qv[j].w, qn))));
  for (int off = 16; off; off >>= 1) qn += __shfl_xor(qn, off);
  qn = fmaxf(sqrtf(qn), EPSV);

  unsigned long long bestkey = 0ull;
  for (int row = wid; row < NMEM; row += nw) {
    const float4* rp = (const float4*)(bank + (size_t)row * DIM);
    // prefetch next row we will touch (global_prefetch_b8)
    int nrow = row + nw;
    if (nrow < NMEM)
      __builtin_prefetch(bank + (size_t)nrow * DIM + lane * 32, 0, 0);

    float dacc = 0.f, nacc = 0.f;
    #pragma unroll
    for (int j = 0; j < 8; ++j) {                           // 8 x b128, fully coalesced
      float4 v = rp[lane + 32 * j];
      dacc = fmaf(v.x, qv[j].x, fmaf(v.y, qv[j].y, fmaf(v.z, qv[j].z, fmaf(v.w, qv[j].w, dacc))));
      nacc = fmaf(v.x, v.x,     fmaf(v.y, v.y,     fmaf(v.z, v.z,     fmaf(v.w, v.w,     nacc))));
    }
    for (int off = 16; off; off >>= 1) {
      dacc += __shfl_xor(dacc, off);
      nacc += __shfl_xor(nacc, off);
    }
    float sim = dacc / (fmaxf(sqrtf(nacc), EPSV) * qn);
    // pack: sim (ordered) in high 32, ~row in low 32 -> max key == argmax, ties -> smaller row
    unsigned long long key =
        ((unsigned long long)f32_to_ordered(sim) << 32) | (unsigned int)(~row);
    bestkey = (key > bestkey) ? key : bestkey;
  }

  __shared__ unsigned long long sb[8];
  if (lane == 0) sb[wib] = bestkey;
  __syncthreads();
  if (threadIdx.x == 0) {
    unsigned long long m = sb[0];
    #pragma unroll
    for (int i = 1; i < 8; ++i) m = (sb[i] > m) ? sb[i] : m;
    atomicMax(best, m);                                     // one u64 atomic per block
  }
}

// ---- K3: decode matvec with V_WMMA_F32_16X16X4_F32 ----------------------
// decoded[d] = sum_k w_dec[d][k] * best_mem[k] + b_dec[d]; zero if sim<=0.65
__global__ void k_decode(const float* __restrict__ bank,
                         const float* __restrict__ w_dec,
                         const float* __restrict__ b_dec,
                         const unsigned long long* __restrict__ best,
                         float* __restrict__ out) {
  const int lane = threadIdx.x;          // one wave32 per block
  const int d0   = blockIdx.x * 16;      // 64 blocks cover 1024 outputs

  unsigned long long key = *best;
  unsigned int idx = ~((unsigned int)key);
  float sim = ordered_to_f32((unsigned int)(key >> 32));
  const float* m = bank + (size_t)idx * DIM;

  // A-tile (16x4 f32) layout: lanes 0-15 -> M=0..15 with K=vgpr; lanes 16-31 -> K=vgpr+2
  const int mrow  = lane & 15;
  const int khalf = (lane >> 4) * 2;
  const float* wrow = w_dec + (size_t)(d0 + mrow) * DIM + khalf;

  v8f c = {};
  for (int k0 = 0; k0 < DIM; k0 += 4) {
    v2f a = *(const v2f*)(wrow + k0);            // w_dec tile (A)
    v2f b = *(const v2f*)(m + k0 + khalf);       // best_mem chunk broadcast to all 16 B columns
    c = __builtin_amdgcn_wmma_f32_16x16x4_f32(
        /*neg_a=*/false, a, /*neg_b=*/false, b,
        /*c_mod=*/(short)0, c, /*reuse_a=*/false, /*reuse_b=*/false);
  }

  bool pass = sim > THRESH;
  // C layout: vgpr v, lanes 0-15 -> M=v ; lanes 16-31 -> M=v+8. All columns identical.
  if (lane == 0 || lane == 16) {
    int rbase = d0 + (lane >> 4) * 8;
    #pragma unroll
    for (int v = 0; v < 8; ++v)
      out[rbase + v] = pass ? (c[v] + b_dec[rbase + v]) : 0.f;
  }
}

extern "C" void kernel_launch(void* const* d_in, const int* in_sizes, int n_in,
                              void* d_out, int out_size, void* d_ws, size_t ws_size,
                              hipStream_t stream) {
  const float* query = (const float*)d_in[0];
  const float* bank  = (const float*)d_in[1];
  const float* w_dec = (const float*)d_in[2];
  const float* b_dec = (const float*)d_in[3];
  float* out = (float*)d_out;

  float* part = (float*)d_ws;                                             // 8*1024 f32
  unsigned long long* best =
      (unsigned long long*)((char*)d_ws + 8 * 1024 * sizeof(float));      // 8B, aligned

  k_partial_q<<<32, 256, 0, stream>>>(query, part, best);
  k_sims<<<1024, 256, 0, stream>>>(bank, part, best);
  k_decode<<<64, 32, 0, stream>>>(bank, w_dec, b_dec, best, out);
}